// KernelizedAttention_54537494724811
// MI455X (gfx1250) — compile-verified
//
#include <hip/hip_runtime.h>
#include <hip/hip_bf16.h>

// Problem constants (from reference): N=4, L=8192, H=8, D=64, DIM=512
#define NB   4
#define LSEQ 8192
#define NH   8
#define HD   64
#define DIM  512
#define CHK  64
#define NCHUNK (LSEQ / CHK)
#define NBH  (NB * NH)
#define SZS  (HD * HD + HD)     // per-chunk state: S (64x64) + z (64) = 4160 floats

typedef __attribute__((ext_vector_type(2))) float v2f;
typedef __attribute__((ext_vector_type(8))) float v8f;

__device__ __forceinline__ v8f wmma4(v2f a, v2f b, v8f c) {
  // V_WMMA_F32_16X16X4_F32 : D = A(16x4) * B(4x16) + C(16x16), fp32
  return __builtin_amdgcn_wmma_f32_16x16x4_f32(
      /*neg_a=*/false, a, /*neg_b=*/false, b,
      /*c_mod=*/(short)0, c, /*reuse_a=*/false, /*reuse_b=*/false);
}

__device__ __forceinline__ float softplus1(float x) {
  return (x > 20.f) ? x : log1pf(__expf(x));
}
__device__ __forceinline__ float softplus2(float x) {
  return softplus1(softplus1(x));
}

// ---------------------------------------------------------------------------
// Pass A — projections: O = epilogue(X @ W^T), written in (N,H,L,D) layout.
// grid = (32768/64, 8, 3); block = 256 (8 waves).
// ---------------------------------------------------------------------------
#define PAD_A 20
#define PAD_B 65

__global__ __launch_bounds__(256)
void proj_kernel(const float* __restrict__ query,
                 const float* __restrict__ key,
                 const float* __restrict__ Wq,
                 const float* __restrict__ Wk,
                 const float* __restrict__ Wv,
                 float* __restrict__ Qf,
                 float* __restrict__ Kf,
                 float* __restrict__ Vf) {
  const int which = blockIdx.z;
  const int h     = blockIdx.y;
  const int mblk  = blockIdx.x;

  const float* __restrict__ X = (which == 0) ? query : key;
  const float* __restrict__ W = (which == 0) ? Wq : (which == 1) ? Wk : Wv;
  float*       __restrict__ O = (which == 0) ? Qf : (which == 1) ? Kf : Vf;

  __shared__ float As[64][PAD_A];        // As[m][k]
  __shared__ float Bs[16][PAD_B];        // Bs[k][n] = W[n0+n][k0+k]

  const int tid  = threadIdx.x;
  const int wave = tid >> 5;
  const int lane = tid & 31;
  const int half = lane >> 4;
  const int l16  = lane & 15;

  const long m0g = (long)mblk * 64;
  const int  n0g = h * 64;

  const int ti = wave & 3;
  const int nj = wave >> 2;

  v8f acc0 = {0.f,0.f,0.f,0.f,0.f,0.f,0.f,0.f};
  v8f acc1 = acc0;

  const int r_ld = tid >> 2;
  const int c_ld = (tid & 3) << 2;

  for (int k0 = 0; k0 < DIM; k0 += 16) {
    const float4 a4 = *(const float4*)&X[(m0g + r_ld) * DIM + k0 + c_ld];
    As[r_ld][c_ld + 0] = a4.x; As[r_ld][c_ld + 1] = a4.y;
    As[r_ld][c_ld + 2] = a4.z; As[r_ld][c_ld + 3] = a4.w;
    const float4 b4 = *(const float4*)&W[(long)(n0g + r_ld) * DIM + k0 + c_ld];
    Bs[c_ld + 0][r_ld] = b4.x; Bs[c_ld + 1][r_ld] = b4.y;
    Bs[c_ld + 2][r_ld] = b4.z; Bs[c_ld + 3][r_ld] = b4.w;
    __syncthreads();

    #pragma unroll
    for (int kk = 0; kk < 4; kk++) {
      const int kb = kk * 4 + half * 2;
      v2f a;  a.x  = As[ti * 16 + l16][kb + 0];
              a.y  = As[ti * 16 + l16][kb + 1];
      v2f b0; b0.x = Bs[kb + 0][nj * 16 + l16];
              b0.y = Bs[kb + 1][nj * 16 + l16];
      v2f b1; b1.x = Bs[kb + 0][(nj + 2) * 16 + l16];
              b1.y = Bs[kb + 1][(nj + 2) * 16 + l16];
      acc0 = wmma4(a, b0, acc0);
      acc1 = wmma4(a, b1, acc1);
    }
    __syncthreads();
  }

  const long nb = m0g / LSEQ;
  const long l0 = m0g % LSEQ;
  float* __restrict__ Obase = O + ((nb * NH + h) * (long)LSEQ + l0) * HD;
  #pragma unroll
  for (int r = 0; r < 8; r++) {
    const int m  = ti * 16 + r + half * 8;
    const int d0 = nj * 16 + l16;
    const int d1 = (nj + 2) * 16 + l16;
    float v0 = acc0[r], v1 = acc1[r];
    if (which != 2) { v0 = softplus2(v0); v1 = softplus2(v1); }
    Obase[(long)m * HD + d0] = v0;
    Obase[(long)m * HD + d1] = v1;
  }
}

// ---------------------------------------------------------------------------
// Pass B — per-chunk stats (fully parallel): Sc = K^T V, zc = colsum(K).
// grid = (128 chunks, 32 bh); block = 256 (8 waves, 2 S-tiles each).
// ---------------------------------------------------------------------------
#define PR 68

__global__ __launch_bounds__(256)
void chunkstat_kernel(const float* __restrict__ Kf,
                      const float* __restrict__ Vf,
                      float* __restrict__ SZ) {
  const int c  = blockIdx.x;
  const int bh = blockIdx.y;

  __shared__ float Ks[CHK][PR];
  __shared__ float Vs[CHK][PR];

  const int tid  = threadIdx.x;
  const int wave = tid >> 5;
  const int lane = tid & 31;
  const int half = lane >> 4;
  const int l16  = lane & 15;

  const float* __restrict__ Kc = Kf + ((size_t)bh * LSEQ + (size_t)c * CHK) * HD;
  const float* __restrict__ Vc = Vf + ((size_t)bh * LSEQ + (size_t)c * CHK) * HD;

  #pragma unroll
  for (int i = tid; i < (CHK * HD) / 4; i += 256) {
    const int r = i >> 4, cc = (i & 15) << 2;
    *(float4*)&Ks[r][cc] = ((const float4*)Kc)[i];
    *(float4*)&Vs[r][cc] = ((const float4*)Vc)[i];
  }
  __syncthreads();

  const int di = wave & 3;
  const int ej = wave >> 2;                  // and ej + 2
  const v8f vzero = {0.f,0.f,0.f,0.f,0.f,0.f,0.f,0.f};
  v8f s0 = vzero, s1 = vzero;

  #pragma unroll
  for (int kt = 0; kt < 4; kt++) {
    #pragma unroll
    for (int kk = 0; kk < 4; kk++) {
      const int t0 = kt * 16 + kk * 4 + half * 2;
      v2f af; af.x = Ks[t0 + 0][di * 16 + l16];   // A[d][t] = K[t][d]
              af.y = Ks[t0 + 1][di * 16 + l16];
      v2f b0; b0.x = Vs[t0 + 0][ej * 16 + l16];
              b0.y = Vs[t0 + 1][ej * 16 + l16];
      v2f b1; b1.x = Vs[t0 + 0][(ej + 2) * 16 + l16];
              b1.y = Vs[t0 + 1][(ej + 2) * 16 + l16];
      s0 = wmma4(af, b0, s0);
      s1 = wmma4(af, b1, s1);
    }
  }

  float* __restrict__ Sc = SZ + ((size_t)bh * NCHUNK + c) * SZS;
  #pragma unroll
  for (int r = 0; r < 8; r++) {
    const int d = di * 16 + r + half * 8;
    Sc[d * HD + ej * 16 + l16]       = s0[r];
    Sc[d * HD + (ej + 2) * 16 + l16] = s1[r];
  }
  if (tid < HD) {
    float zz = 0.f;
    for (int t = 0; t < CHK; t++) zz += Ks[t][tid];
    Sc[HD * HD + tid] = zz;
  }
}

// ---------------------------------------------------------------------------
// Pass C — exclusive prefix scan of chunk states along the sequence (in place).
// grid = 32 (one per bh); block = 256. Each thread owns 16 strided S elements.
// The only sequential pass left: ~130 MB traffic, trivially HBM-bound.
// ---------------------------------------------------------------------------
__global__ __launch_bounds__(256)
void scan_kernel(float* __restrict__ SZ) {
  const int bh  = blockIdx.x;
  const int tid = threadIdx.x;

  float run[16];
  #pragma unroll
  for (int j = 0; j < 16; j++) run[j] = 0.f;
  float rz = 0.f;

  float* __restrict__ base = SZ + (size_t)bh * NCHUNK * SZS;
  for (int c = 0; c < NCHUNK; c++) {
    float* __restrict__ p = base + (size_t)c * SZS;
    if (c + 1 < NCHUNK) {                        // hide latency on serial chain
      __builtin_prefetch(p + SZS + tid, 0, 0);
      __builtin_prefetch(p + SZS + 2048 + tid, 0, 0);
    }
    float cur[16];
    #pragma unroll
    for (int j = 0; j < 16; j++) cur[j] = p[j * 256 + tid];
    const float cz = (tid < HD) ? p[HD * HD + tid] : 0.f;
    #pragma unroll
    for (int j = 0; j < 16; j++) { p[j * 256 + tid] = run[j]; run[j] += cur[j]; }
    if (tid < HD) p[HD * HD + tid] = rz;
    rz += cz;
  }
}

// ---------------------------------------------------------------------------
// Pass D — outputs (fully parallel): A = mask(Q K^T);
//   out = (A V + Q S_prev) / (rowsum(A) + Q . z_prev)
// grid = (128 chunks, 32 bh); block = 256.
// ---------------------------------------------------------------------------
__global__ __launch_bounds__(256)
void attn_out_kernel(const float* __restrict__ Qf,
                     const float* __restrict__ Kf,
                     const float* __restrict__ Vf,
                     const float* __restrict__ SZ,
                     float* __restrict__ out) {
  const int c  = blockIdx.x;
  const int bh = blockIdx.y;
  const int nb = bh >> 3;
  const int h  = bh & 7;

  __shared__ float Qs[CHK][PR];
  __shared__ float Ks[CHK][PR];
  __shared__ float Vs[CHK][PR];
  __shared__ float Am[CHK][PR];
  __shared__ float Sp[HD][PR];
  __shared__ float zp[HD];
  __shared__ float den[CHK];

  const int tid  = threadIdx.x;
  const int wave = tid >> 5;
  const int lane = tid & 31;
  const int half = lane >> 4;
  const int l16  = lane & 15;

  const float* __restrict__ Qc = Qf + ((size_t)bh * LSEQ + (size_t)c * CHK) * HD;
  const float* __restrict__ Kc = Kf + ((size_t)bh * LSEQ + (size_t)c * CHK) * HD;
  const float* __restrict__ Vc = Vf + ((size_t)bh * LSEQ + (size_t)c * CHK) * HD;
  const float* __restrict__ Sc = SZ + ((size_t)bh * NCHUNK + c) * SZS;

  #pragma unroll
  for (int i = tid; i < (CHK * HD) / 4; i += 256) {
    const int r = i >> 4, cc = (i & 15) << 2;
    *(float4*)&Qs[r][cc] = ((const float4*)Qc)[i];
    *(float4*)&Ks[r][cc] = ((const float4*)Kc)[i];
    *(float4*)&Vs[r][cc] = ((const float4*)Vc)[i];
    *(float4*)&Sp[r][cc] = ((const float4*)Sc)[i];
  }
  if (tid < HD) zp[tid] = Sc[HD * HD + tid];
  __syncthreads();

  const v8f vzero = {0.f,0.f,0.f,0.f,0.f,0.f,0.f,0.f};

  // ---- A = mask(Q K^T), lower-triangular tiles only ----
  #pragma unroll
  for (int pick = 0; pick < 2; pick++) {
    const int id = wave + pick * 8;              // wave-uniform
    const int ti = id >> 2, tj = id & 3;
    if (tj > ti) continue;
    v8f a = vzero;
    #pragma unroll
    for (int k = 0; k < 16; k++) {
      const int kb = k * 4 + half * 2;
      v2f af; af.x = Qs[ti * 16 + l16][kb + 0];
              af.y = Qs[ti * 16 + l16][kb + 1];
      v2f bf; bf.x = Ks[tj * 16 + l16][kb + 0];
              bf.y = Ks[tj * 16 + l16][kb + 1];
      a = wmma4(af, bf, a);
    }
    #pragma unroll
    for (int r = 0; r < 8; r++) {
      const int m = ti * 16 + r + half * 8;
      const int s = tj * 16 + l16;
      Am[m][s] = (s > m) ? 0.f : a[r];
    }
  }
  __syncthreads();

  // ---- acc = A V + Q S_prev ; den = rowsum(A) + Q . z_prev ----
  const int ti_o = wave & 3;
  const int ej0  = wave >> 2;
  v8f acc0 = vzero, acc1 = vzero;

  for (int k = 0; k <= ti_o; k++) {
    #pragma unroll
    for (int kk = 0; kk < 4; kk++) {
      const int kb = k * 16 + kk * 4 + half * 2;
      v2f af; af.x = Am[ti_o * 16 + l16][kb + 0];
              af.y = Am[ti_o * 16 + l16][kb + 1];
      v2f b0; b0.x = Vs[kb + 0][ej0 * 16 + l16];
              b0.y = Vs[kb + 1][ej0 * 16 + l16];
      v2f b1; b1.x = Vs[kb + 0][(ej0 + 2) * 16 + l16];
              b1.y = Vs[kb + 1][(ej0 + 2) * 16 + l16];
      acc0 = wmma4(af, b0, acc0);
      acc1 = wmma4(af, b1, acc1);
    }
  }
  #pragma unroll
  for (int k = 0; k < 4; k++) {
    #pragma unroll
    for (int kk = 0; kk < 4; kk++) {
      const int kb = k * 16 + kk * 4 + half * 2;
      v2f af; af.x = Qs[ti_o * 16 + l16][kb + 0];
              af.y = Qs[ti_o * 16 + l16][kb + 1];
      v2f b0; b0.x = Sp[kb + 0][ej0 * 16 + l16];
              b0.y = Sp[kb + 1][ej0 * 16 + l16];
      v2f b1; b1.x = Sp[kb + 0][(ej0 + 2) * 16 + l16];
              b1.y = Sp[kb + 1][(ej0 + 2) * 16 + l16];
      acc0 = wmma4(af, b0, acc0);
      acc1 = wmma4(af, b1, acc1);
    }
  }
  if (tid < CHK) {
    float d = 0.f;
    for (int e = 0; e < HD; e++) d += Qs[tid][e] * zp[e];
    for (int s = 0; s <= tid; s++) d += Am[tid][s];
    den[tid] = d;
  }
  __syncthreads();

  // ---- out = acc / den, (N, L, DIM) layout ----
  const long rowbase = (long)nb * LSEQ + (long)c * CHK;
  #pragma unroll
  for (int r = 0; r < 8; r++) {
    const int t = ti_o * 16 + r + half * 8;
    const float dn = den[t];
    out[(rowbase + t) * DIM + h * HD + ej0 * 16 + l16]       = acc0[r] / dn;
    out[(rowbase + t) * DIM + h * HD + (ej0 + 2) * 16 + l16] = acc1[r] / dn;
  }
}

extern "C" void kernel_launch(void* const* d_in, const int* in_sizes, int n_in,
                              void* d_out, int out_size, void* d_ws, size_t ws_size,
                              hipStream_t stream) {
  (void)in_sizes; (void)n_in; (void)out_size; (void)ws_size;

  const float* query = (const float*)d_in[0];
  const float* key   = (const float*)d_in[1];
  const float* Wq    = (const float*)d_in[2];
  const float* Wk    = (const float*)d_in[3];
  const float* Wv    = (const float*)d_in[4];
  float* out = (float*)d_out;

  const size_t feat_elems = (size_t)NB * LSEQ * DIM;      // 16M floats each
  float* Qf = (float*)d_ws;
  float* Kf = Qf + feat_elems;
  float* Vf = Kf + feat_elems;
  float* SZ = Vf + feat_elems;                            // 32*128*4160 floats

  // Pass A: fused projections (WMMA fp32 GEMM + double-softplus epilogue)
  dim3 pgrid((NB * LSEQ) / 64, NH, 3);
  proj_kernel<<<pgrid, 256, 0, stream>>>(query, key, Wq, Wk, Wv, Qf, Kf, Vf);

  // Pass B: per-chunk K^T V + colsum(K), fully parallel over 4096 blocks
  dim3 bgrid(NCHUNK, NBH);
  chunkstat_kernel<<<bgrid, 256, 0, stream>>>(Kf, Vf, SZ);

  // Pass C: exclusive prefix scan of states (only sequential pass; HBM-bound)
  scan_kernel<<<NBH, 256, 0, stream>>>(SZ);

  // Pass D: outputs, fully parallel over 4096 blocks
  attn_out_kernel<<<bgrid, 256, 0, stream>>>(Qf, Kf, Vf, SZ, out);
}